// RNN_23003844837547
// MI455X (gfx1250) — compile-verified
//
#include <hip/hip_runtime.h>
#include <math.h>

#define SEQ_LEN 100000
#define NH 256                     // INPUT_SIZE == HIDDEN_SIZE == 256

typedef __attribute__((ext_vector_type(2))) float v2f;
typedef __attribute__((ext_vector_type(8))) float v8f;

#ifndef __has_builtin
#define __has_builtin(x) 0
#endif
#if __has_builtin(__builtin_amdgcn_wmma_f32_16x16x4_f32)
#define HAVE_WMMA_F32X4 1
#else
#define HAVE_WMMA_F32X4 0
#endif

// ---------------------------------------------------------------------------
// Kernel 2 (emitted FIRST so the disasm snippet shows its inner loop):
// sequential scan h_t = tanh(xi_t + W h_{t-1}), single workgroup.
// Thread j owns output row j and keeps W[j][0..255] ENTIRELY IN VGPRs
// (128 x v2f = 256 VGPRs; gfx1250 wave32 allows up to 1024 VGPRs/wave, so
// 8 waves x ~280 VGPRs fits at 2 waves/SIMD). LDS is used only for the tiny
// double-buffered h vector (2 KB): broadcast ds reads per step. The inner
// product targets packed V_PK_FMA_F32 (2 MAC/lane/clk) with 4 rotating
// accumulators to break the FMA latency chain. One __syncthreads per step;
// xi_{t+1} prefetched from global one step ahead. xi is consumed from
// d_out[t] and the same slot is overwritten with h_t (column-owned).
// ---------------------------------------------------------------------------
__global__ __launch_bounds__(256) void rnn_scan(const float* __restrict__ hh_w,
                                                float* __restrict__ out) {
  __shared__ float hbuf[2][NH];
  const int j = threadIdx.x;              // this thread owns output column j

  // Pull this thread's weight row into registers (one-time, L2-resident).
  v2f wr[NH / 2];
#pragma unroll
  for (int k2 = 0; k2 < NH / 2; ++k2)
    wr[k2] = *(const v2f*)(hh_w + (size_t)j * NH + 2 * k2);

  hbuf[0][j] = 0.0f;                      // h_{-1} = 0
  __syncthreads();

  float xin  = out[j];                    // prefetched xi_0
  float hnew = 0.0f;
  for (int t = 0; t < SEQ_LEN; ++t) {
    const float xi_c = xin;
    if (t + 1 < SEQ_LEN) xin = out[(size_t)(t + 1) * NH + j];  // prefetch xi_{t+1}
    const float* hr = hbuf[t & 1];
    float*       hw = hbuf[(t & 1) ^ 1];

    v2f a0 = {xi_c, 0.f}, a1 = {0.f, 0.f}, a2 = {0.f, 0.f}, a3 = {0.f, 0.f};
#pragma unroll
    for (int k8 = 0; k8 < NH / 8; ++k8) {             // 32 iters, 8 k each
      float4 hA = *(const float4*)(hr + 8 * k8);      // broadcast b128
      float4 hB = *(const float4*)(hr + 8 * k8 + 4);  // broadcast b128
      v2f h0 = {hA.x, hA.y}, h1 = {hA.z, hA.w};
      v2f h2 = {hB.x, hB.y}, h3 = {hB.z, hB.w};
      a0 = __builtin_elementwise_fma(wr[4 * k8 + 0], h0, a0);
      a1 = __builtin_elementwise_fma(wr[4 * k8 + 1], h1, a1);
      a2 = __builtin_elementwise_fma(wr[4 * k8 + 2], h2, a2);
      a3 = __builtin_elementwise_fma(wr[4 * k8 + 3], h3, a3);
    }
    v2f s = (a0 + a1) + (a2 + a3);
    hnew = tanhf(s.x + s.y);

    out[(size_t)t * NH + j] = hnew;       // overwrite xi_t with h_t
    hw[j] = hnew;
    __syncthreads();                      // one barrier per step
  }
  out[(size_t)SEQ_LEN * NH + j] = hnew;   // trailing hid output
}

// ---------------------------------------------------------------------------
// Kernel 1: xi[t,h] = sum_i x[t,i] * hi_w[h,i], written straight into d_out.
// One wave32 per 16(t) x 16(h) output tile, K=256 in 64 steps of
// V_WMMA_F32_16X16X4_F32 (full fp32 precision on the matrix pipe).
// Verified in rounds 1-2: clean wmma chain, b64 load clauses, no spills.
// ---------------------------------------------------------------------------
#if HAVE_WMMA_F32X4
__global__ __launch_bounds__(256) void xi_gemm_wmma(const float* __restrict__ x,
                                                    const float* __restrict__ hi_w,
                                                    float* __restrict__ out) {
  const int wave   = (blockIdx.x * 256 + threadIdx.x) >> 5;  // 100000 waves
  const int lane   = threadIdx.x & 31;
  const int t_tile = wave >> 4;          // 6250 t-tiles
  const int h_tile = wave & 15;          // 16 h-tiles
  const int t0 = t_tile << 4;
  const int h0 = h_tile << 4;

  // f32 16x16x4 operand layout: lane L -> row L%16, K = 2*(L/16) + {0,1}
  const int r  = lane & 15;
  const int kc = (lane >> 4) << 1;       // 0 or 2
  const float* ap = x    + (size_t)(t0 + r) * NH + kc;   // A = x tile (t x i)
  const float* bp = hi_w + (size_t)(h0 + r) * NH + kc;   // B = hi_w^T tile (i x h)

  v8f c = {0.f, 0.f, 0.f, 0.f, 0.f, 0.f, 0.f, 0.f};
#pragma unroll 8
  for (int kb = 0; kb < NH; kb += 4) {
    v2f a = *(const v2f*)(ap + kb);
    v2f b = *(const v2f*)(bp + kb);
    c = __builtin_amdgcn_wmma_f32_16x16x4_f32(false, a, false, b,
                                              (short)0, c, false, false);
  }

  // D layout: VGPR v -> row (v + 8*(lane/16)), col (lane%16)
  const int mbase = (lane >> 4) << 3;
  float* op = out + (size_t)(t0 + mbase) * NH + h0 + r;
#pragma unroll
  for (int v = 0; v < 8; ++v) op[(size_t)v * NH] = c[v];
}
#else
// Scalar fallback (keeps build green if the f32-WMMA builtin is absent).
__global__ __launch_bounds__(256) void xi_gemm_naive(const float* __restrict__ x,
                                                     const float* __restrict__ hi_w,
                                                     float* __restrict__ out) {
  __shared__ float xs[NH];
  const int t = blockIdx.x;
  const int j = threadIdx.x;
  xs[j] = x[(size_t)t * NH + j];
  __syncthreads();
  float acc = 0.f;
#pragma unroll 8
  for (int i = 0; i < NH; ++i) acc = fmaf(hi_w[(size_t)j * NH + i], xs[i], acc);
  out[(size_t)t * NH + j] = acc;
}
#endif

// ---------------------------------------------------------------------------
extern "C" void kernel_launch(void* const* d_in, const int* in_sizes, int n_in,
                              void* d_out, int out_size, void* d_ws, size_t ws_size,
                              hipStream_t stream) {
  const float* x    = (const float*)d_in[0];   // (100000, 1, 256) f32
  const float* hi_w = (const float*)d_in[1];   // (256, 256) f32
  const float* hh_w = (const float*)d_in[2];   // (256, 256) f32
  float* out = (float*)d_out;                  // 100000*256 + 256 f32

#if HAVE_WMMA_F32X4
  // 6250 t-tiles * 16 h-tiles = 100000 waves; 8 waves per 256-thread block.
  xi_gemm_wmma<<<(SEQ_LEN / 16) * 16 / 8, 256, 0, stream>>>(x, hi_w, out);
#else
  xi_gemm_naive<<<SEQ_LEN, 256, 0, stream>>>(x, hi_w, out);
#endif
  rnn_scan<<<1, 256, 0, stream>>>(hh_w, out);
}